// DeepSeekV3Attention_56126632624299
// MI455X (gfx1250) — compile-verified
//
#include <hip/hip_runtime.h>
#include <hip/hip_bf16.h>

// ---------------------------------------------------------------------------
// DeepSeek-V3 MLA attention forward, fp32, CDNA5 (gfx1250).
// All matmuls run on V_WMMA_F32_16X16X4_F32 (native fp32 matrix path).
// GEMMs use a double-buffered LDS pipeline + global_prefetch.
// ---------------------------------------------------------------------------

typedef __attribute__((ext_vector_type(2))) float v2f;
typedef __attribute__((ext_vector_type(8))) float v8f;

#define H_   32
#define DN_  128
#define DR_  64
#define DV_  128
#define HID_ 4096
#define QR_  1536
#define KVR_ 512
#define SEQ_ 2048
#define TOK_ 4096        // 2 * 2048 flattened tokens
#define DQK_ 192         // DN + DR

__device__ __forceinline__ v8f wmma4(v2f a, v2f b, v8f c) {
  // D = A(16x4,f32) * B(4x16,f32) + C(16x16,f32)
  return __builtin_amdgcn_wmma_f32_16x16x4_f32(
      /*neg_a=*/false, a, /*neg_b=*/false, b,
      /*c_mod=*/(short)0, c, /*reuse_a=*/false, /*reuse_b=*/false);
}

// ---------------------------------------------------------------------------
// Generic GEMM: C[M,N] = A[M,K] * B[N,K]^T  (weights stored row-major [N,K]).
// Block = 256 threads (8 waves), 128x128 output tile.
// Ping-pong LDS: next K-tile's global loads overlap the current tile's WMMAs.
// Requires M%128==0, N%128==0, K%16==0 (true for every projection here).
// ---------------------------------------------------------------------------
#define GBM 128
#define GBN 128
#define GBK 16
#define GPAD 20   // LDS row pitch (floats): 80B, 16B-aligned, breaks bank collisions

__global__ __launch_bounds__(256)
void gemm_nt_wmma(const float* __restrict__ A, const float* __restrict__ B,
                  float* __restrict__ C, int M, int N, int K) {
  __shared__ float As[2][GBM][GPAD];
  __shared__ float Bs[2][GBN][GPAD];

  const int tid  = threadIdx.x;
  const int wave = tid >> 5;
  const int lane = tid & 31;
  const int l16  = lane & 15;
  const int lh   = lane >> 4;
  const int wm   = wave >> 1;              // 0..3 -> 32-row group
  const int wn   = wave & 1;               // 0..1 -> 64-col group
  const int bm   = blockIdx.y * GBM;
  const int bn   = blockIdx.x * GBN;

  // Staging slots: 512 float4 per 128x16 tile; this thread owns rows sr and
  // sr+64 at column sc (same column for both).
  const int sr = tid >> 2;                 // 0..63
  const int sc = (tid & 3) << 2;           // 0,4,8,12
  const float* Ap = A + (size_t)(bm + sr) * K + sc;
  const float* Bp = B + (size_t)(bn + sr) * K + sc;
  const size_t half = (size_t)64 * K;

  const v8f vz = {0.f,0.f,0.f,0.f,0.f,0.f,0.f,0.f};
  v8f acc[2][4];
#pragma unroll
  for (int i = 0; i < 2; ++i)
#pragma unroll
    for (int j = 0; j < 4; ++j) acc[i][j] = vz;

  // Preload first K-tile into LDS buffer 0.
  {
    float4 ra0 = *(const float4*)(Ap);
    float4 ra1 = *(const float4*)(Ap + half);
    float4 rb0 = *(const float4*)(Bp);
    float4 rb1 = *(const float4*)(Bp + half);
    *(float4*)&As[0][sr][sc]      = ra0;
    *(float4*)&As[0][sr + 64][sc] = ra1;
    *(float4*)&Bs[0][sr][sc]      = rb0;
    *(float4*)&Bs[0][sr + 64][sc] = rb1;
  }
  __syncthreads();

  int buf = 0;
  for (int k0 = 0; k0 < K; k0 += GBK) {
    const int kn = k0 + GBK;
    float4 ra0, ra1, rb0, rb1;
    if (kn < K) {
      // Issue next tile's global loads; they retire under the WMMA stream.
      ra0 = *(const float4*)(Ap + kn);
      ra1 = *(const float4*)(Ap + half + kn);
      rb0 = *(const float4*)(Bp + kn);
      rb1 = *(const float4*)(Bp + half + kn);
    }
    if (kn + GBK < K) {
      // Pre-warm L2 two tiles ahead (global_prefetch_b8).
      __builtin_prefetch(Ap + kn + GBK, 0, 0);
      __builtin_prefetch(Ap + half + kn + GBK, 0, 0);
      __builtin_prefetch(Bp + kn + GBK, 0, 0);
      __builtin_prefetch(Bp + half + kn + GBK, 0, 0);
    }

    // Compute on the current buffer.
#pragma unroll
    for (int kk = 0; kk < GBK; kk += 4) {
      v2f a[2], b[4];
#pragma unroll
      for (int i = 0; i < 2; ++i)
        a[i] = *(const v2f*)&As[buf][wm * 32 + i * 16 + l16][kk + 2 * lh];
#pragma unroll
      for (int j = 0; j < 4; ++j)
        b[j] = *(const v2f*)&Bs[buf][wn * 64 + j * 16 + l16][kk + 2 * lh];
#pragma unroll
      for (int i = 0; i < 2; ++i)
#pragma unroll
        for (int j = 0; j < 4; ++j)
          acc[i][j] = wmma4(a[i], b[j], acc[i][j]);
    }

    if (kn < K) {
      *(float4*)&As[buf ^ 1][sr][sc]      = ra0;
      *(float4*)&As[buf ^ 1][sr + 64][sc] = ra1;
      *(float4*)&Bs[buf ^ 1][sr][sc]      = rb0;
      *(float4*)&Bs[buf ^ 1][sr + 64][sc] = rb1;
    }
    __syncthreads();
    buf ^= 1;
  }

#pragma unroll
  for (int i = 0; i < 2; ++i)
#pragma unroll
    for (int j = 0; j < 4; ++j) {
      int m0 = bm + wm * 32 + i * 16 + 8 * lh;   // C/D layout: row = r + 8*half
      int n  = bn + wn * 64 + j * 16 + l16;      //             col = lane%16
#pragma unroll
      for (int r = 0; r < 8; ++r)
        C[(size_t)(m0 + r) * N + n] = acc[i][j][r];
    }
}

// ---------------------------------------------------------------------------
// In-place RMSNorm over rows of length D. One 256-thread block per row.
// ---------------------------------------------------------------------------
__global__ __launch_bounds__(256)
void rms_norm_inplace(float* __restrict__ x, const float* __restrict__ w, int D) {
  const int row = blockIdx.x;
  float* xr = x + (size_t)row * D;
  float s = 0.f;
  for (int i = threadIdx.x; i < D; i += 256) { float v = xr[i]; s += v * v; }
#pragma unroll
  for (int off = 16; off; off >>= 1) s += __shfl_xor(s, off, 32);
  __shared__ float red[8];
  if ((threadIdx.x & 31) == 0) red[threadIdx.x >> 5] = s;
  __syncthreads();
  float tot = 0.f;
#pragma unroll
  for (int i = 0; i < 8; ++i) tot += red[i];
  const float inv = rsqrtf(tot / (float)D + 1e-6f);
  for (int i = threadIdx.x; i < D; i += 256) xr[i] = xr[i] * inv * w[i];
}

// ---------------------------------------------------------------------------
// RoPE (concat[f,f] tables + rotate_half), in place. One wave per (token,head).
// x row = token, head h occupies [h*hs + off, h*hs + off + 64).
// ---------------------------------------------------------------------------
__global__ __launch_bounds__(32)
void rope_kernel(float* __restrict__ x, const int* __restrict__ pos,
                 int hs, int off) {
  const int t = blockIdx.x;
  const int h = blockIdx.y;
  const int i = threadIdx.x;                 // 0..31
  const float p = (float)pos[t];
  const float invf = __expf(-9.210340371976184f * (float)i / 32.0f); // 10000^(-i/32)
  float sn, cs;
  __sincosf(p * invf, &sn, &cs);
  float* base = x + (size_t)t * (H_ * hs) + (size_t)h * hs + off;
  const float x1 = base[i];
  const float x2 = base[i + 32];
  base[i]      = x1 * cs - x2 * sn;
  base[i + 32] = x2 * cs + x1 * sn;
}

// ---------------------------------------------------------------------------
// Flash-style MLA attention (no mask, full softmax over 2048 keys).
// Grid: (q_tiles=32, heads=32, batch=2). Block: 128 threads = 4 waves,
// each wave owns 16 query rows. K tile = [k_nope | roped k_rope] (64x192).
// ---------------------------------------------------------------------------
__global__ __launch_bounds__(128)
void mla_attn(const float* __restrict__ Qb,   // [TOK, H*192] (nope|rope, roped)
              const float* __restrict__ KVb,  // [TOK, H*256] (k_nope|v)
              const float* __restrict__ KRb,  // [TOK, H*64]  (roped)
              float* __restrict__ Ctx) {      // [TOK, H*128]
  const int h  = blockIdx.y;
  const int b  = blockIdx.z;
  const int q0 = blockIdx.x * 64;
  const int tid  = threadIdx.x;
  const int wave = tid >> 5;
  const int lane = tid & 31;
  const int l16  = lane & 15;
  const int lh   = lane >> 4;

  __shared__ float Qs[64][DQK_ + 4];   // pitch 196 floats -> bank step 4
  __shared__ float Ks[64][DQK_ + 4];
  __shared__ float Vs[64][DV_ + 4];
  __shared__ float Ps[4][16][64 + 4];  // per-wave probability tile

  const size_t rowQ  = (size_t)H_ * DQK_;   // 6144
  const size_t rowKV = (size_t)H_ * 256;    // 8192
  const size_t rowKR = (size_t)H_ * DR_;    // 2048
  const size_t rowC  = (size_t)H_ * DV_;    // 4096

  // Stage Q tile once: 64 x 192 floats = 3072 float4.
  const float* Qp = Qb + (size_t)(b * SEQ_ + q0) * rowQ + h * DQK_;
  for (int c = tid; c < 64 * 48; c += 128) {
    int r = c / 48, cc = (c % 48) * 4;
    *(float4*)&Qs[r][cc] = *(const float4*)(Qp + (size_t)r * rowQ + cc);
  }

  const v8f vz = {0.f,0.f,0.f,0.f,0.f,0.f,0.f,0.f};
  v8f accO[8];
#pragma unroll
  for (int t = 0; t < 8; ++t) accO[t] = vz;
  float row_max[8], row_sum[8];
#pragma unroll
  for (int r = 0; r < 8; ++r) { row_max[r] = -1e30f; row_sum[r] = 0.f; }

  const float scale = 0.07216878364870323f;   // 1/sqrt(192)

  for (int k0 = 0; k0 < SEQ_; k0 += 64) {
    __syncthreads();   // previous iteration done reading Ks/Vs (also covers Qs staging)
    const float* Kn = KVb + (size_t)(b * SEQ_ + k0) * rowKV + h * 256;
    const float* Vp = Kn + 128;
    const float* Kr = KRb + (size_t)(b * SEQ_ + k0) * rowKR + h * DR_;
    for (int c = tid; c < 64 * 32; c += 128) {      // 64 x 128 floats, twice
      int r = c >> 5, cc = (c & 31) << 2;
      *(float4*)&Ks[r][cc] = *(const float4*)(Kn + (size_t)r * rowKV + cc);
      *(float4*)&Vs[r][cc] = *(const float4*)(Vp + (size_t)r * rowKV + cc);
    }
    for (int c = tid; c < 64 * 16; c += 128) {      // 64 x 64 rope floats
      int r = c >> 4, cc = (c & 15) << 2;
      *(float4*)&Ks[r][DN_ + cc] = *(const float4*)(Kr + (size_t)r * rowKR + cc);
    }
    __syncthreads();

    // ---- S = Q K^T for this wave's 16 rows x 64 keys -----------------------
    v8f s[4];
#pragma unroll
    for (int j = 0; j < 4; ++j) s[j] = vz;
    for (int kk = 0; kk < DQK_; kk += 4) {
      v2f a = *(const v2f*)&Qs[wave * 16 + l16][kk + 2 * lh];
#pragma unroll
      for (int j = 0; j < 4; ++j) {
        v2f bb = *(const v2f*)&Ks[j * 16 + l16][kk + 2 * lh];
        s[j] = wmma4(a, bb, s[j]);
      }
    }

    // ---- online softmax (rows live in 16-lane halves) ----------------------
    float nm[8], corr[8], psum[8];
#pragma unroll
    for (int r = 0; r < 8; ++r) {
      float m = fmaxf(fmaxf(s[0][r], s[1][r]), fmaxf(s[2][r], s[3][r])) * scale;
#pragma unroll
      for (int off = 8; off; off >>= 1) m = fmaxf(m, __shfl_xor(m, off, 16));
      nm[r]   = fmaxf(row_max[r], m);
      corr[r] = __expf(row_max[r] - nm[r]);
      row_max[r] = nm[r];
      psum[r] = 0.f;
    }
#pragma unroll
    for (int j = 0; j < 4; ++j)
#pragma unroll
      for (int r = 0; r < 8; ++r) {
        float p = __expf(s[j][r] * scale - nm[r]);
        Ps[wave][r + 8 * lh][j * 16 + l16] = p;
        psum[r] += p;
      }
#pragma unroll
    for (int r = 0; r < 8; ++r) {
      float ps = psum[r];
#pragma unroll
      for (int off = 8; off; off >>= 1) ps += __shfl_xor(ps, off, 16);
      row_sum[r] = row_sum[r] * corr[r] + ps;
    }
#pragma unroll
    for (int t = 0; t < 8; ++t)
#pragma unroll
      for (int r = 0; r < 8; ++r) accO[t][r] *= corr[r];

    asm volatile("s_wait_dscnt 0" ::: "memory");   // wave-private P tile visible

    // ---- O += P V ----------------------------------------------------------
    for (int kk = 0; kk < 64; kk += 4) {
      v2f a = *(const v2f*)&Ps[wave][l16][kk + 2 * lh];
#pragma unroll
      for (int t = 0; t < 8; ++t) {
        v2f bb;
        bb[0] = Vs[kk + 2 * lh][t * 16 + l16];
        bb[1] = Vs[kk + 2 * lh + 1][t * 16 + l16];
        accO[t] = wmma4(a, bb, accO[t]);
      }
    }
  }

  // ---- normalize and write context [TOK, H*DV] -----------------------------
  float* Cp = Ctx + (size_t)(b * SEQ_ + q0 + wave * 16) * rowC + h * DV_;
#pragma unroll
  for (int t = 0; t < 8; ++t)
#pragma unroll
    for (int r = 0; r < 8; ++r) {
      int m = r + 8 * lh;
      Cp[(size_t)m * rowC + t * 16 + l16] = accO[t][r] / row_sum[r];
    }
}

// ---------------------------------------------------------------------------
// Host orchestration.
// Inputs: 0 hidden, 1 position_ids, 2 q_a_w, 3 q_a_ln_w, 4 q_b_w, 5 kv_a_w,
//         6 kv_a_ln_w, 7 kv_b_w, 8 k_rope_w, 9 o_w
// ---------------------------------------------------------------------------
extern "C" void kernel_launch(void* const* d_in, const int* in_sizes, int n_in,
                              void* d_out, int out_size, void* d_ws, size_t ws_size,
                              hipStream_t stream) {
  const float* hidden   = (const float*)d_in[0];
  const int*   pos      = (const int*)d_in[1];
  const float* q_a_w    = (const float*)d_in[2];
  const float* q_a_ln   = (const float*)d_in[3];
  const float* q_b_w    = (const float*)d_in[4];
  const float* kv_a_w   = (const float*)d_in[5];
  const float* kv_a_ln  = (const float*)d_in[6];
  const float* kv_b_w   = (const float*)d_in[7];
  const float* k_rope_w = (const float*)d_in[8];
  const float* o_w      = (const float*)d_in[9];
  float* out = (float*)d_out;

  float* w = (float*)d_ws;
  size_t o = 0;
  float* qa  = w + o; o += (size_t)TOK_ * QR_;            // [4096,1536]
  float* q   = w + o; o += (size_t)TOK_ * (H_ * DQK_);    // [4096,6144]
  float* kva = w + o; o += (size_t)TOK_ * KVR_;           // [4096,512]
  float* kv  = w + o; o += (size_t)TOK_ * (H_ * 256);     // [4096,8192]
  float* kr  = w + o; o += (size_t)TOK_ * (H_ * DR_);     // [4096,2048]
  float* ctx = w + o; o += (size_t)TOK_ * (H_ * DV_);     // [4096,4096]

  const dim3 blk256(256);

  // q_a: [4096,4096] x [1536,4096]^T
  gemm_nt_wmma<<<dim3(QR_ / GBN, TOK_ / GBM), blk256, 0, stream>>>(
      hidden, q_a_w, qa, TOK_, QR_, HID_);
  rms_norm_inplace<<<dim3(TOK_), blk256, 0, stream>>>(qa, q_a_ln, QR_);
  // q_b: [4096,1536] x [6144,1536]^T
  gemm_nt_wmma<<<dim3(H_ * DQK_ / GBN, TOK_ / GBM), blk256, 0, stream>>>(
      qa, q_b_w, q, TOK_, H_ * DQK_, QR_);

  // kv_a: [4096,4096] x [512,4096]^T
  gemm_nt_wmma<<<dim3(KVR_ / GBN, TOK_ / GBM), blk256, 0, stream>>>(
      hidden, kv_a_w, kva, TOK_, KVR_, HID_);
  rms_norm_inplace<<<dim3(TOK_), blk256, 0, stream>>>(kva, kv_a_ln, KVR_);
  // kv_b: [4096,512] x [8192,512]^T
  gemm_nt_wmma<<<dim3(H_ * 256 / GBN, TOK_ / GBM), blk256, 0, stream>>>(
      kva, kv_b_w, kv, TOK_, H_ * 256, KVR_);

  // k_rope: [4096,4096] x [2048,4096]^T
  gemm_nt_wmma<<<dim3(H_ * DR_ / GBN, TOK_ / GBM), blk256, 0, stream>>>(
      hidden, k_rope_w, kr, TOK_, H_ * DR_, HID_);

  // RoPE in place (q rope slice at offset 128 of each 192-wide head; kr whole)
  rope_kernel<<<dim3(TOK_, H_), dim3(32), 0, stream>>>(q, pos, DQK_, DN_);
  rope_kernel<<<dim3(TOK_, H_), dim3(32), 0, stream>>>(kr, pos, DR_, 0);

  // attention -> ctx [4096, 4096]
  mla_attn<<<dim3(SEQ_ / 64, H_, 2), dim3(128), 0, stream>>>(q, kv, kr, ctx);

  // o_proj: [4096,4096] x [4096,4096]^T -> out
  gemm_nt_wmma<<<dim3(HID_ / GBN, TOK_ / GBM), blk256, 0, stream>>>(
      ctx, o_w, out, TOK_, HID_, H_ * DV_);
}